// Encoder_74182675137046
// MI455X (gfx1250) — compile-verified
//
#include <hip/hip_runtime.h>

typedef __attribute__((ext_vector_type(2))) float v2f;
typedef __attribute__((ext_vector_type(8))) float v8f;

#define HH 1024
#define WW 1024
#define DD 128
#define NN (HH * WW)

// ws layout (as float*):
//  [0,512)    M = W1@W2  (4 x 128, row-major)
//  [512,640)  c = b1@W2 + b2
//  [640,768)  sum accumulator (128)
//  at float offset 768, as int[5]: rmin, rmax, cmin, cmax, count

// ---------------- Kernel 1: precompute M, c; init scratch ----------------
__global__ __launch_bounds__(128)
void k1_prep(const float* __restrict__ W1, const float* __restrict__ b1,
             const float* __restrict__ W2, const float* __restrict__ b2,
             float* __restrict__ ws) {
  int j = threadIdx.x;  // 0..127 output column
  float m0 = 0.f, m1 = 0.f, m2 = 0.f, m3 = 0.f, cc = 0.f;
  for (int k = 0; k < DD; ++k) {
    float w2 = W2[k * DD + j];
    m0 = fmaf(W1[0 * DD + k], w2, m0);
    m1 = fmaf(W1[1 * DD + k], w2, m1);
    m2 = fmaf(W1[2 * DD + k], w2, m2);
    m3 = fmaf(W1[3 * DD + k], w2, m3);
    cc = fmaf(b1[k], w2, cc);
  }
  ws[0 * DD + j] = m0;
  ws[1 * DD + j] = m1;
  ws[2 * DD + j] = m2;
  ws[3 * DD + j] = m3;
  ws[512 + j] = cc + b2[j];
  ws[640 + j] = 0.f;   // zero sum accumulator every call (graph-replay safe)
  if (j == 0) {
    int* st = (int*)(ws + 768);
    st[0] = 0x7fffffff;  // rmin
    st[1] = -1;          // rmax
    st[2] = 0x7fffffff;  // cmin
    st[3] = -1;          // cmax
    st[4] = 0;           // count
  }
}

// ---------------- Kernel 2: masked row/col min/max + count ----------------
__global__ __launch_bounds__(256)
void k2_stats(const float* __restrict__ x0, int* __restrict__ st) {
  int tid = blockIdx.x * blockDim.x + threadIdx.x;
  int stride = gridDim.x * blockDim.x;
  int rmn = 0x7fffffff, rmx = -1, cmn = 0x7fffffff, cmx = -1, cnt = 0;
  for (int p = tid; p < NN; p += stride) {
    if (x0[p] != 0.f) {
      int r = p >> 10, c = p & 1023;
      rmn = min(rmn, r); rmx = max(rmx, r);
      cmn = min(cmn, c); cmx = max(cmx, c);
      ++cnt;
    }
  }
  // wave32 reduction
  for (int off = 16; off; off >>= 1) {
    rmn = min(rmn, __shfl_xor(rmn, off, 32));
    rmx = max(rmx, __shfl_xor(rmx, off, 32));
    cmn = min(cmn, __shfl_xor(cmn, off, 32));
    cmx = max(cmx, __shfl_xor(cmx, off, 32));
    cnt += __shfl_xor(cnt, off, 32);
  }
  if ((threadIdx.x & 31) == 0) {
    atomicMin(&st[0], rmn);
    atomicMax(&st[1], rmx);
    atomicMin(&st[2], cmn);
    atomicMax(&st[3], cmx);
    atomicAdd(&st[4], cnt);
  }
}

// ---------------- Kernel 3: main fused WMMA kernel ----------------
// Each wave processes tiles of 16 consecutive pixels.
// A (16x4 f32): lanes 0-15 hold {nr,nc} (K=0,1), lanes 16-31 hold {x0,x1} (K=2,3).
// B (4x16 f32): column slab of M; lane n%16 holds the K pair per lane half.
// C: bias broadcast. D -> relu (v_med3) -> masked weighted row-sum accumulate.
__global__ __launch_bounds__(256)
void k3_main(const float* __restrict__ x, const float* __restrict__ ws,
             float* __restrict__ wsum) {
  const float* Mm = ws;
  const float* cvec = ws + 512;
  const int* st = (const int*)(ws + 768);
  __shared__ float lsum[DD];

  int t = threadIdx.x;
  if (t < DD) lsum[t] = 0.f;
  __syncthreads();

  int lane = t & 31;
  int khalf = (lane >> 4) & 1;  // 0: K=0,1 ; 1: K=2,3
  int sub = lane & 15;          // pixel-in-tile / matrix column
  int shbase = khalf * 8;       // this lane's rows are v + shbase

  // scalar (SGPR) wave id -> uniform loop, no exec-mask churn, WMMA-safe
  int wvs = __builtin_amdgcn_readfirstlane(t >> 5);

  float rmin = (float)st[0], rmax = (float)st[1];
  float cmin = (float)st[2], cmax = (float)st[3];
  float invr = 1.f / (rmax - rmin);
  float invc = 1.f / (cmax - cmin);

  // Load B slabs (M columns) and bias-broadcast C once per wave.
  v2f Bm[8];
  v8f cin[8];
#pragma unroll
  for (int tt = 0; tt < 8; ++tt) {
    int col = tt * 16 + sub;
    Bm[tt].x = Mm[(khalf * 2 + 0) * DD + col];
    Bm[tt].y = Mm[(khalf * 2 + 1) * DD + col];
    float b = cvec[col];
#pragma unroll
    for (int v = 0; v < 8; ++v) cin[tt][v] = b;
  }

  float acc[8];
#pragma unroll
  for (int tt = 0; tt < 8; ++tt) acc[tt] = 0.f;

  const float* x0 = x;
  const float* x1 = x + NN;
  int gw = blockIdx.x * 8 + wvs;
  int nw = gridDim.x * 8;
  const int ntiles = NN / 16;
  const float pinf = __builtin_inff();

  for (int tile = gw; tile < ntiles; tile += nw) {
    int p = tile * 16 + sub;
    float v0 = x0[p];
    float v1 = x1[p];

    // whole-wave mask bits: bit i = pixel (base+i) nonzero (lanes 16-31 dup)
    unsigned mball = (unsigned)__ballot(v0 != 0.f);
    unsigned mb = mball >> shbase;

    // branchless A operand
    float rr = (float)(p >> 10);
    float cl = (float)(p & 1023);
    float nr = (rr - rmin) * invr;
    float nc = (cl - cmin) * invc;
    v2f a;
    a.x = khalf ? v0 : nr;
    a.y = khalf ? v1 : nc;

    // row weights for this lane (0/1 floats), shared by all 8 column tiles
    float wrow[8];
#pragma unroll
    for (int v = 0; v < 8; ++v)
      wrow[v] = (float)((mb >> v) & 1u);

#pragma unroll
    for (int tt = 0; tt < 8; ++tt) {
      v8f d = __builtin_amdgcn_wmma_f32_16x16x4_f32(
          false, a, false, Bm[tt], (short)0, cin[tt], false, false);
      float s = 0.f;
#pragma unroll
      for (int v = 0; v < 8; ++v) {
        // relu in a single v_med3_f32 (no maxnum canonicalization)
        float r = __builtin_amdgcn_fmed3f(d[v], 0.f, pinf);
        s = fmaf(wrow[v], r, s);
      }
      acc[tt] += s;
    }
  }

  // workgroup reduction: lane l and l+16 both hold column tt*16+sub
#pragma unroll
  for (int tt = 0; tt < 8; ++tt)
    atomicAdd(&lsum[tt * 16 + sub], acc[tt]);
  __syncthreads();
  if (t < DD) atomicAdd(&wsum[t], lsum[t]);
}

// ---------------- Kernel 4: out = (sum/count) @ W3 + b3 ----------------
__global__ __launch_bounds__(128)
void k4_final(const float* __restrict__ ws, const float* __restrict__ W3,
              const float* __restrict__ b3, float* __restrict__ out) {
  __shared__ float avg[DD];
  int j = threadIdx.x;
  const int* st = (const int*)(ws + 768);
  float inv = 1.f / (float)st[4];
  avg[j] = ws[640 + j] * inv;
  __syncthreads();
  float s = b3[j];
  for (int k = 0; k < DD; ++k)
    s = fmaf(avg[k], W3[k * DD + j], s);
  out[j] = s;
}

extern "C" void kernel_launch(void* const* d_in, const int* in_sizes, int n_in,
                              void* d_out, int out_size, void* d_ws, size_t ws_size,
                              hipStream_t stream) {
  (void)in_sizes; (void)n_in; (void)out_size; (void)ws_size;
  const float* x  = (const float*)d_in[0];  // (2, H, W)
  const float* W1 = (const float*)d_in[1];  // (4, 128)
  const float* b1 = (const float*)d_in[2];  // (128,)
  const float* W2 = (const float*)d_in[3];  // (128, 128)
  const float* b2 = (const float*)d_in[4];  // (128,)
  const float* W3 = (const float*)d_in[5];  // (128, 128)
  const float* b3 = (const float*)d_in[6];  // (128,)
  float* out = (float*)d_out;               // (128,)
  float* ws = (float*)d_ws;
  int* st = (int*)(ws + 768);

  k1_prep<<<1, 128, 0, stream>>>(W1, b1, W2, b2, ws);
  k2_stats<<<256, 256, 0, stream>>>(x, st);
  k3_main<<<512, 256, 0, stream>>>(x, ws, ws + 640);
  k4_final<<<1, 128, 0, stream>>>(ws, W3, b3, out);
}